// EdgeConvMemoryEfficient_77790447665154
// MI455X (gfx1250) — compile-verified
//
#include <hip/hip_runtime.h>

typedef float v2f __attribute__((ext_vector_type(2)));
typedef float v8f __attribute__((ext_vector_type(8)));

#define CDIM 64
#define NPTS 4096
#define KNN  20
#define BATCH 4
#define EDGES (BATCH * NPTS * KNN)

// ---------------------------------------------------------------------------
// K0: transpose x (B,C,N) -> xt (B,N,C) and compute sq[b,n] = sum_c x^2
// ---------------------------------------------------------------------------
__global__ __launch_bounds__(256)
void prep_kernel(const float* __restrict__ x, float* __restrict__ xt,
                 float* __restrict__ sq) {
  const int b = blockIdx.y;
  const int n = blockIdx.x * blockDim.x + threadIdx.x;
  const float* xb = x + (size_t)b * CDIM * NPTS;
  float* xto = xt + ((size_t)b * NPTS + n) * CDIM;
  float acc = 0.0f;
#pragma unroll 8
  for (int c = 0; c < CDIM; ++c) {
    float v = xb[(size_t)c * NPTS + n];   // coalesced across threads in n
    acc += v * v;
    xto[c] = v;
  }
  sq[(size_t)b * NPTS + n] = acc;
}

// ---------------------------------------------------------------------------
// Kw: wsum = Wc - Wd, wd = Wd  (W is (O, 2C), Wc = W[:, :C], Wd = W[:, C:])
//     also zero the BN stats accumulators (must happen every launch)
// ---------------------------------------------------------------------------
__global__ __launch_bounds__(256)
void wprep_kernel(const float* __restrict__ W, float* __restrict__ wsum,
                  float* __restrict__ wd, float* __restrict__ stats) {
  const int i = blockIdx.x * blockDim.x + threadIdx.x;  // 0..4095
  const int o = i >> 6, c = i & 63;
  const float wc  = W[o * (2 * CDIM) + c];
  const float wdv = W[o * (2 * CDIM) + CDIM + c];
  wsum[i] = wc - wdv;
  wd[i]   = wdv;
  if (i < 128) stats[i] = 0.0f;
}

// ---------------------------------------------------------------------------
// helper: load a 16-row x 64-col fp32 tile in WMMA f32 16x16x4 operand order.
// lane l -> point row (base + l%16); half = l/16 selects K pair within step.
// For step ks, operand holds elements c = 4*ks + 2*half + {0,1}.
// A (16xK) and B (Kx16, = transposed point tile) use the same per-lane layout.
// ---------------------------------------------------------------------------
__device__ __forceinline__ void load_tile16x64(const float* __restrict__ base,
                                               int m, int h, v2f (&t)[16]) {
  const float* p = base + (size_t)m * CDIM + 2 * h;
#pragma unroll
  for (int ks = 0; ks < 16; ++ks)
    t[ks] = *(const v2f*)(p + 4 * ks);
}

// ---------------------------------------------------------------------------
// K1: fused pairwise-distance (WMMA fp32) + on-the-fly top-k (k=20 smallest).
// One wave per 16 query rows; loops over all 256 column tiles.
// d2 is finished in registers from the WMMA accumulator; a per-row threshold
// (20th-best so far, kept in LDS) filters candidates wave-wide; rare
// survivors go through a small LDS queue to the owning lane's register list.
// Stale thresholds are conservative (only admit extras), so this is exact.
// ---------------------------------------------------------------------------
__global__ __launch_bounds__(32)
void knn_topk_kernel(const float* __restrict__ xt, const float* __restrict__ sq,
                     int* __restrict__ idxout) {
  const int b = blockIdx.y;
  const int rowBase = blockIdx.x * 16;
  const int lane = threadIdx.x;
  const int m = lane & 15;
  const int h = lane >> 4;

  const float* xtb = xt + (size_t)b * NPTS * CDIM;
  const float* sqb = sq + (size_t)b * NPTS;

  v2f a[16];
  load_tile16x64(xtb + (size_t)rowBase * CDIM, m, h, a);

  float sqr[8];  // |x|^2 of this lane's 8 accumulator rows (loop-invariant)
#pragma unroll
  for (int j = 0; j < 8; ++j) sqr[j] = sqb[rowBase + j + 8 * h];

  float bestd[KNN];
  int besti[KNN];
#pragma unroll
  for (int i = 0; i < KNN; ++i) { bestd[i] = 3.0e38f; besti[i] = 0; }

  __shared__ float ldsThr[16];       // per-row 20th-best d2 so far
  __shared__ float qd[256];          // candidate queue: d2
  __shared__ int qc[256];            // candidate queue: (col<<4)|row
  __shared__ unsigned int qn;

  if (lane < 16) ldsThr[lane] = 3.0e38f;
  if (lane == 0) qn = 0u;
  __syncthreads();

  for (int ct = 0; ct < NPTS / 16; ++ct) {
    const int colBase = ct * 16;

    v2f bt[16];
    load_tile16x64(xtb + (size_t)colBase * CDIM, m, h, bt);
    const float sqc = sqb[colBase + m];  // |x|^2 of this lane's column

    v8f acc = {};
#pragma unroll
    for (int ks = 0; ks < 16; ++ks)
      acc = __builtin_amdgcn_wmma_f32_16x16x4_f32(
          false, a[ks], false, bt[ks], (short)0, acc, false, false);

    // wave-wide filter: all 32 lanes test their 8 d2 values vs row threshold
#pragma unroll
    for (int j = 0; j < 8; ++j) {
      const int row = j + 8 * h;
      float d2 = sqr[j] + sqc - 2.0f * acc[j];
      d2 = fmaxf(d2, 0.0f);  // relu; sqrt is monotone -> rank by d2
      if (d2 < ldsThr[row]) {
        const unsigned slot = atomicAdd(&qn, 1u);
        qd[slot] = d2;
        qc[slot] = ((colBase + m) << 4) | row;
      }
    }
    __syncthreads();  // pushes visible; queue stable

    const int qcount = (int)qn;
    if (lane == 0) qn = 0u;  // same wave: LDS in-order, read above already done
    if (lane < 16 && qcount > 0) {
      for (int q = 0; q < qcount; ++q) {
        const int e = qc[q];
        if ((e & 15) == lane) {  // this owner's row
          const float d2 = qd[q];
          if (d2 < bestd[KNN - 1]) {
            int jj = KNN - 1;
            while (jj > 0 && bestd[jj - 1] > d2) {  // strict: stable on ties
              bestd[jj] = bestd[jj - 1];
              besti[jj] = besti[jj - 1];
              --jj;
            }
            bestd[jj] = d2;
            besti[jj] = e >> 4;
          }
        }
      }
      ldsThr[lane] = bestd[KNN - 1];
    }
    __syncthreads();  // thr/queue updates ordered before next tile's pushes
  }

  if (lane < 16) {
    int* op = idxout + ((size_t)b * NPTS + rowBase + lane) * KNN;
#pragma unroll
    for (int i = 0; i < KNN; ++i) op[i] = besti[i];
  }
}

// ---------------------------------------------------------------------------
// K2: y1 = xt @ (Wc-Wd)^T, y2 = xt @ Wd^T via WMMA fp32 (B,N,O)
// ---------------------------------------------------------------------------
__global__ __launch_bounds__(32)
void ygemm_kernel(const float* __restrict__ xt, const float* __restrict__ wsum,
                  const float* __restrict__ wd, float* __restrict__ y1,
                  float* __restrict__ y2) {
  const int b = blockIdx.y;
  const int rowBase = blockIdx.x * 16;
  const int lane = threadIdx.x;
  const int m = lane & 15;
  const int h = lane >> 4;
  const float* xtb = xt + (size_t)b * NPTS * CDIM;

  v2f a[16];
  load_tile16x64(xtb + (size_t)rowBase * CDIM, m, h, a);

#pragma unroll
  for (int mat = 0; mat < 2; ++mat) {
    const float* Wm = (mat == 0) ? wsum : wd;
    float* y = (mat == 0) ? y1 : y2;
#pragma unroll
    for (int ot = 0; ot < 4; ++ot) {
      v2f bt[16];
      load_tile16x64(Wm + (size_t)(ot * 16) * CDIM, m, h, bt);  // B[k][o]=Wm[o][k]
      v8f acc = {};
#pragma unroll
      for (int ks = 0; ks < 16; ++ks)
        acc = __builtin_amdgcn_wmma_f32_16x16x4_f32(
            false, a[ks], false, bt[ks], (short)0, acc, false, false);
#pragma unroll
      for (int j = 0; j < 8; ++j) {
        const int row = rowBase + j + 8 * h;
        y[((size_t)b * NPTS + row) * 64 + ot * 16 + m] = acc[j];
      }
    }
  }
}

// ---------------------------------------------------------------------------
// K3: BN statistics: sum / sumsq of h over all (b,n,k) per channel o
//     h[b,n,k,o] = y1[b,n,o] + y2[b,idx[b,n,k],o]
// ---------------------------------------------------------------------------
__global__ __launch_bounds__(256)
void stats_kernel(const float* __restrict__ y1, const float* __restrict__ y2,
                  const int* __restrict__ nbr, float* __restrict__ stats) {
  const int o = threadIdx.x & 63;
  const int grp = threadIdx.x >> 6;  // 0..3
  float s = 0.0f, ss = 0.0f;
  for (int e = blockIdx.x * 4 + grp; e < EDGES; e += gridDim.x * 4) {
    const int pn = e / KNN;      // flat b*N + n
    const int b = pn >> 12;      // N = 4096
    const int nb = nbr[e];
    const float hv = y1[(size_t)pn * 64 + o] +
                     y2[(((size_t)b << 12) + nb) * 64 + o];
    s += hv;
    ss += hv * hv;
  }
  __shared__ float red[512];
  red[threadIdx.x] = s;
  red[256 + threadIdx.x] = ss;
  __syncthreads();
  if (threadIdx.x < 64) {
    float ts = 0.0f, tss = 0.0f;
#pragma unroll
    for (int g = 0; g < 4; ++g) {
      ts += red[g * 64 + o];
      tss += red[256 + g * 64 + o];
    }
    atomicAdd(&stats[o], ts);
    atomicAdd(&stats[64 + o], tss);
  }
}

// ---------------------------------------------------------------------------
// K4: per-channel affine: scale = gamma*rsqrt(var+eps), bias = beta - mean*scale
// ---------------------------------------------------------------------------
__global__ __launch_bounds__(64)
void finalize_kernel(const float* __restrict__ stats,
                     const float* __restrict__ gamma,
                     const float* __restrict__ beta, float* __restrict__ sb) {
  const int o = threadIdx.x;
  const float inv = 1.0f / (float)EDGES;
  const float mean = stats[o] * inv;
  const float var = stats[64 + o] * inv - mean * mean;
  const float scale = gamma[o] * rsqrtf(var + 1e-5f);
  sb[o] = scale;
  sb[64 + o] = beta[o] - mean * scale;
}

// ---------------------------------------------------------------------------
// K5: out[b,o,n] = max_k lrelu(h*scale + bias)
// ---------------------------------------------------------------------------
__global__ __launch_bounds__(256)
void out_kernel(const float* __restrict__ y1, const float* __restrict__ y2,
                const int* __restrict__ nbr, const float* __restrict__ sb,
                float* __restrict__ out) {
  const int o = threadIdx.x & 63;
  const int pt = blockIdx.x * 4 + (threadIdx.x >> 6);  // flat b*N + n
  const int b = pt >> 12;
  const int n = pt & 4095;
  const float scale = sb[o], bias = sb[64 + o];
  const float y1v = y1[(size_t)pt * 64 + o];
  const int* np = nbr + (size_t)pt * KNN;
  float mx = -3.0e38f;
#pragma unroll 4
  for (int k = 0; k < KNN; ++k) {
    const int nb = np[k];
    const float hv = y1v + y2[(((size_t)b << 12) + nb) * 64 + o];
    float v = hv * scale + bias;
    v = (v >= 0.0f) ? v : 0.2f * v;
    mx = fmaxf(mx, v);
  }
  out[((size_t)b * 64 + o) * NPTS + n] = mx;
}

// ---------------------------------------------------------------------------
extern "C" void kernel_launch(void* const* d_in, const int* in_sizes, int n_in,
                              void* d_out, int out_size, void* d_ws,
                              size_t ws_size, hipStream_t stream) {
  const float* x     = (const float*)d_in[0];  // (B,C,N)
  const float* W     = (const float*)d_in[1];  // (O,2C)
  const float* gamma = (const float*)d_in[2];  // (O,)
  const float* beta  = (const float*)d_in[3];  // (O,)
  // d_in[4] = k (==20, compile-time KNN)

  float* ws    = (float*)d_ws;
  float* xt    = ws;                 // B*N*C      = 1,048,576 f
  float* sq    = xt + 1048576;       // B*N        =    16,384 f
  float* wsum  = sq + 16384;         // O*C        =     4,096 f
  float* wd    = wsum + 4096;        // O*C        =     4,096 f
  float* y1    = wd + 4096;          // B*N*O      = 1,048,576 f
  float* y2    = y1 + 1048576;       // B*N*O      = 1,048,576 f
  float* stats = y2 + 1048576;       // 128 f (sum, sumsq)
  float* sb    = stats + 128;        // 128 f (scale, bias)
  int*   nbr   = (int*)(sb + 128);   // B*N*K      =   327,680 i32
  float* out   = (float*)d_out;      // (B,O,N)

  prep_kernel<<<dim3(NPTS / 256, BATCH), 256, 0, stream>>>(x, xt, sq);
  wprep_kernel<<<16, 256, 0, stream>>>(W, wsum, wd, stats);
  knn_topk_kernel<<<dim3(NPTS / 16, BATCH), 32, 0, stream>>>(xt, sq, nbr);
  ygemm_kernel<<<dim3(NPTS / 16, BATCH), 32, 0, stream>>>(xt, wsum, wd, y1, y2);
  stats_kernel<<<1024, 256, 0, stream>>>(y1, y2, nbr, stats);
  finalize_kernel<<<1, 64, 0, stream>>>(stats, gamma, beta, sb);
  out_kernel<<<BATCH * NPTS / 4, 256, 0, stream>>>(y1, y2, nbr, sb, out);
}